// ARMA_Net_17617955848509
// MI455X (gfx1250) — compile-verified
//
#include <hip/hip_runtime.h>
#include <cmath>

// ---------------------------------------------------------------------------
// ARMA GCN (2 stacks, 2 layers per conv) for MI455X / gfx1250, wave32.
// Dense GEMMs -> V_WMMA_F32_16X16X4_F32 (fp32 matrix path), branch-free
// inner loop: one global_load_b64 (A frag) + one ds_load_b64 (B frag) per step.
// Sparse propagation -> edge-parallel gather/scale + global_atomic_add_f32;
// gather tables (51MB) are L2-resident on the 192MB L2 -> L2-BW bound.
// ---------------------------------------------------------------------------

typedef __attribute__((ext_vector_type(2))) float v2f;
typedef __attribute__((ext_vector_type(8))) float v8f;

#define CDIV(a, b) (((a) + (b) - 1) / (b))

static const int kN = 100000;
static const int kE = 1600000;
static const int kFin = 100;
static const int kH = 64;
static const int kC = 18;
static const int kCPad = 20;  // kC rounded up to multiple of 4 (WMMA K-step)
static const int kK = 2;      // stacks

// ---------------------------------------------------------------------------
// elementwise helpers
// ---------------------------------------------------------------------------
__global__ void fill_kernel(float* __restrict__ p, int n, float v) {
    int i = blockIdx.x * blockDim.x + threadIdx.x;
    if (i < n) p[i] = v;
}

__global__ void deg_kernel(const long long* __restrict__ ei, float* __restrict__ deg, int E) {
    int e = blockIdx.x * blockDim.x + threadIdx.x;
    if (e < E) {
        int t = (int)ei[(size_t)E + e];
        unsafeAtomicAdd(&deg[t], 1.0f);
    }
}

__global__ void dis_kernel(const float* __restrict__ deg, float* __restrict__ dis, int n) {
    int i = blockIdx.x * blockDim.x + threadIdx.x;
    if (i < n) {
        float d = deg[i];
        dis[i] = (d > 0.0f) ? rsqrtf(fmaxf(d, 1.0f)) : 0.0f;
    }
}

__global__ void norm_kernel(const long long* __restrict__ ei, const float* __restrict__ dis,
                            float* __restrict__ nrm, int E) {
    int e = blockIdx.x * blockDim.x + threadIdx.x;
    if (e < E) {
        int s = (int)ei[e];
        int t = (int)ei[(size_t)E + e];
        nrm[e] = dis[s] * dis[t];
    }
}

// ---------------------------------------------------------------------------
// fp32 WMMA GEMM: Out[k] = A[k] (n x Kd, row stride lda) * W[k] (Kd x Fout)
// Requirements (guaranteed by caller): lda % 4 == 0, KdPad <= lda, so the
// A fragment is an unconditional aligned 8-byte load. Row index is clamped
// (OOB rows compute garbage, stores are guarded). Weight tile is staged
// TRANSPOSED and zero-padded in LDS: wlds[ff][kk], so B frag = ds_load_b64.
// blockIdx.y = stack k. 256 threads = 8 waves; wave (rt,ct) owns 16x16 of D.
// Fragment layout per CDNA5 ISA 7.12.2:
//   A 16x4 : lane M = lane&15; reg0 = A[M][k0 + 2*(lane>>4)], reg1 = +1
//   B 4x16 : lane N = lane&15; reg0 = B[k0 + 2*(lane>>4)][N], reg1 = +1
//   D 16x16: lane N = lane&15; VGPR j = D[ j + 8*(lane>>4) ][N]
// ---------------------------------------------------------------------------
__global__ void gemm_wmma_f32(const float* __restrict__ A, size_t aStrideK, int lda,
                              const float* __restrict__ W,
                              float* __restrict__ Out,
                              int n, int Kd, int Fout) {
    extern __shared__ float wlds[];   // [FoutPad][KdPad], transposed, zero-padded
    const int k = blockIdx.y;
    const int FoutPad = (Fout + 15) & ~15;
    const int KdPad   = (Kd + 3) & ~3;
    const float* Ak = A + (size_t)k * aStrideK;
    const float* Wk = W + (size_t)k * (size_t)Kd * (size_t)Fout;
    float* Outk     = Out + (size_t)k * (size_t)n * (size_t)Fout;

    // stage transposed, zero-padded weights into LDS
    const int tot = KdPad * FoutPad;
    for (int i = threadIdx.x; i < tot; i += blockDim.x) {
        int ff = i / KdPad;
        int kk = i - ff * KdPad;
        wlds[i] = (kk < Kd && ff < Fout) ? Wk[(size_t)kk * Fout + ff] : 0.0f;
    }
    __syncthreads();

    const int wave = threadIdx.x >> 5;
    const int lane = threadIdx.x & 31;
    const int Ct = FoutPad >> 4;   // col tiles (1,2,4 for our shapes)
    const int Rt = 8 / Ct;         // row tiles per block
    const int rt = wave / Ct;
    const int ct = wave - rt * Ct;
    if (rt >= Rt) return;

    const int rowbase = blockIdx.x * (Rt << 4) + (rt << 4);
    const int row = min(rowbase + (lane & 15), n - 1);   // clamp: always in-bounds
    const int col = (ct << 4) + (lane & 15);
    const int kh  = (lane >> 4) << 1;

    const float* arow = Ak + (size_t)row * lda + kh;          // 8B aligned
    const float* bcol = wlds + (size_t)col * KdPad + kh;      // 8B aligned

    v8f acc = {0.f, 0.f, 0.f, 0.f, 0.f, 0.f, 0.f, 0.f};
    for (int k0 = 0; k0 < KdPad; k0 += 4) {
        v2f a = *reinterpret_cast<const v2f*>(arow + k0);     // global_load_b64
        v2f b = *reinterpret_cast<const v2f*>(bcol + k0);     // ds_load_b64
        acc = __builtin_amdgcn_wmma_f32_16x16x4_f32(
            /*neg_a=*/false, a, /*neg_b=*/false, b,
            /*c_mod=*/(short)0, acc, /*reuse_a=*/false, /*reuse_b=*/false);
    }

    if (col < Fout) {
        const int mrow = rowbase + ((lane >> 4) << 3);
#pragma unroll
        for (int j = 0; j < 8; ++j) {
            int r = mrow + j;
            if (r < n) Outk[(size_t)r * Fout + col] = acc[j];
        }
    }
}

// ---------------------------------------------------------------------------
// Edge-parallel propagation: agg[k][tgt] += norm[e] * h[k][src]
// vec4 version for F multiple of 4 (conv1, F=64): 1 thread = 1 edge x 4 feats
// ---------------------------------------------------------------------------
__global__ void scatter_vec4_kernel(const float* __restrict__ h,
                                    const long long* __restrict__ ei,
                                    const float* __restrict__ nrm,
                                    float* __restrict__ agg,
                                    int E, int n, int F) {
    const int chunks = F >> 2;
    int t = blockIdx.x * blockDim.x + threadIdx.x;
    if (t >= E * chunks) return;
    const int e = t / chunks;
    const int c = t - e * chunks;
    const int k = blockIdx.y;
    const int s  = (int)ei[e];
    const int tg = (int)ei[(size_t)E + e];
    const float w = nrm[e];
    const float4 hv = *reinterpret_cast<const float4*>(
        &h[((size_t)k * n + s) * F + (c << 2)]);
    float* dst = &agg[((size_t)k * n + tg) * F + (c << 2)];
    unsafeAtomicAdd(dst + 0, hv.x * w);
    unsafeAtomicAdd(dst + 1, hv.y * w);
    unsafeAtomicAdd(dst + 2, hv.z * w);
    unsafeAtomicAdd(dst + 3, hv.w * w);
}

// scalar version for arbitrary F (conv2, F=18)
__global__ void scatter_scalar_kernel(const float* __restrict__ h,
                                      const long long* __restrict__ ei,
                                      const float* __restrict__ nrm,
                                      float* __restrict__ agg,
                                      int E, int n, int F) {
    int t = blockIdx.x * blockDim.x + threadIdx.x;
    if (t >= E * F) return;
    const int e = t / F;
    const int f = t - e * F;
    const int k = blockIdx.y;
    const int s  = (int)ei[e];
    const int tg = (int)ei[(size_t)E + e];
    const float v = h[((size_t)k * n + s) * F + f] * nrm[e];
    unsafeAtomicAdd(&agg[((size_t)k * n + tg) * F + f], v);
}

// out[k][n][f] = relu(agg + root + b[k][f]); output row stride may be padded
__global__ void combine_kernel(const float* __restrict__ agg,
                               const float* __restrict__ root,
                               const float* __restrict__ b,
                               float* __restrict__ out,
                               int n, int F, int outStride, int total) {
    int i = blockIdx.x * blockDim.x + threadIdx.x;
    if (i >= total) return;
    const int nf = n * F;
    const int k  = i / nf;
    const int r  = (i - k * nf) / F;
    const int f  = i - k * nf - r * F;
    out[((size_t)k * n + r) * outStride + f] =
        fmaxf(agg[i] + root[i] + b[k * F + f], 0.0f);
}

// x2[i] = relu(mean over stacks)
__global__ void mean_relu_kernel(const float* __restrict__ in, float* __restrict__ out, int nf) {
    int i = blockIdx.x * blockDim.x + threadIdx.x;
    if (i < nf) out[i] = fmaxf(0.5f * (in[i] + in[(size_t)nf + i]), 0.0f);
}

// mean over stacks + log_softmax over C, one node per thread
__global__ void mean_logsoftmax_kernel(const float* __restrict__ in, float* __restrict__ out,
                                       int n, int C) {
    int node = blockIdx.x * blockDim.x + threadIdx.x;
    if (node >= n) return;
    float v[32];
    const size_t off0 = (size_t)node * C;
    const size_t off1 = (size_t)n * C + off0;
    float m = -3.402823466e38f;
    for (int c = 0; c < C; ++c) {
        v[c] = 0.5f * (in[off0 + c] + in[off1 + c]);
        m = fmaxf(m, v[c]);
    }
    float s = 0.0f;
    for (int c = 0; c < C; ++c) s += __expf(v[c] - m);
    const float ls = m + logf(s);
    for (int c = 0; c < C; ++c) out[off0 + c] = v[c] - ls;
}

// ---------------------------------------------------------------------------
// host orchestration
// ---------------------------------------------------------------------------
extern "C" void kernel_launch(void* const* d_in, const int* in_sizes, int n_in,
                              void* d_out, int out_size, void* d_ws, size_t ws_size,
                              hipStream_t stream) {
    (void)in_sizes; (void)n_in; (void)out_size; (void)ws_size;

    const float*     x        = (const float*)d_in[0];
    const long long* ei       = (const long long*)d_in[1];
    const float*     init_w1  = (const float*)d_in[2];
    const float*     w1       = (const float*)d_in[3];
    const float*     root_w1  = (const float*)d_in[4];
    const float*     b1       = (const float*)d_in[5];
    const float*     init_w2  = (const float*)d_in[6];
    const float*     w2       = (const float*)d_in[7];
    const float*     root_w2  = (const float*)d_in[8];
    const float*     b2       = (const float*)d_in[9];
    float* out = (float*)d_out;

    // workspace carve (floats)
    float* ws = (float*)d_ws;
    size_t off = 0;
    float* deg  = ws + off; off += kN;
    float* dis  = ws + off; off += kN;
    float* nrm  = ws + off; off += kE;
    const size_t knh = (size_t)kK * kN * kH;     // 12.8M floats
    float* rootb = ws + off; off += knh;         // root1 / root2 (aliased)
    float* bufA  = ws + off; off += knh;         // h ping
    float* bufB  = ws + off; off += knh;         // out pong
    float* agg   = ws + off; off += knh;         // aggregation
    float* x2    = ws + off; off += (size_t)kN * kH;
    // conv2 t0 activation, K-padded to stride 20, aliases x2 region after use
    float* bufPad = x2;                          // K*N*20 = 4M floats <= 6.4M

    const int TB = 256;
    const size_t knc    = (size_t)kK * kN * kC;
    const size_t kncPad = (size_t)kK * kN * kCPad;

    // ---- gcn norm -------------------------------------------------------
    fill_kernel<<<CDIV(kN, TB), TB, 0, stream>>>(deg, kN, 0.0f);
    deg_kernel<<<CDIV(kE, TB), TB, 0, stream>>>(ei, deg, kE);
    dis_kernel<<<CDIV(kN, TB), TB, 0, stream>>>(deg, dis, kN);
    norm_kernel<<<CDIV(kE, TB), TB, 0, stream>>>(ei, dis, nrm, kE);

    auto gemm = [&](const float* A, size_t aStrideK, int lda, const float* W, float* Out,
                    int n, int Kd, int Fout) {
        const int FoutPad = (Fout + 15) & ~15;
        const int KdPad   = (Kd + 3) & ~3;
        const int Ct = FoutPad >> 4;
        const int Rt = 8 / Ct;
        const int rowsPerBlock = Rt << 4;
        dim3 grid(CDIV(n, rowsPerBlock), kK);
        size_t shmem = (size_t)KdPad * FoutPad * sizeof(float);
        gemm_wmma_f32<<<grid, 256, shmem, stream>>>(A, aStrideK, lda, W, Out, n, Kd, Fout);
    };

    // ---- conv1 (Fin=100 -> H=64) ---------------------------------------
    gemm(x, 0, kFin, init_w1, bufA, kN, kFin, kH);       // h0 = x @ init_w1[k]
    gemm(x, 0, kFin, root_w1, rootb, kN, kFin, kH);      // root1

    // t = 0
    fill_kernel<<<CDIV((int)knh, TB), TB, 0, stream>>>(agg, (int)knh, 0.0f);
    scatter_vec4_kernel<<<dim3(CDIV(kE * (kH / 4), TB), kK), TB, 0, stream>>>(
        bufA, ei, nrm, agg, kE, kN, kH);
    combine_kernel<<<CDIV((int)knh, TB), TB, 0, stream>>>(
        agg, rootb, b1, bufB, kN, kH, kH, (int)knh);
    // t = 1
    gemm(bufB, (size_t)kN * kH, kH, w1, bufA, kN, kH, kH);   // h1 = out[k] @ w1[k]
    fill_kernel<<<CDIV((int)knh, TB), TB, 0, stream>>>(agg, (int)knh, 0.0f);
    scatter_vec4_kernel<<<dim3(CDIV(kE * (kH / 4), TB), kK), TB, 0, stream>>>(
        bufA, ei, nrm, agg, kE, kN, kH);
    combine_kernel<<<CDIV((int)knh, TB), TB, 0, stream>>>(
        agg, rootb, b1, bufB, kN, kH, kH, (int)knh);
    // relu(mean over stacks)
    mean_relu_kernel<<<CDIV(kN * kH, TB), TB, 0, stream>>>(bufB, x2, kN * kH);

    // ---- conv2 (H=64 -> C=18), buffers aliased onto conv1 regions ------
    gemm(x2, 0, kH, init_w2, bufA, kN, kH, kC);        // h0 = x2 @ init_w2[k]
    gemm(x2, 0, kH, root_w2, rootb, kN, kH, kC);       // root2
    // (x2 consumed by the two gemms above; its region now becomes bufPad)

    // t = 0 : combine writes K-padded (stride 20) activation for the w2 GEMM
    fill_kernel<<<CDIV((int)knc, TB), TB, 0, stream>>>(agg, (int)knc, 0.0f);
    scatter_scalar_kernel<<<dim3(CDIV(kE * kC, TB), kK), TB, 0, stream>>>(
        bufA, ei, nrm, agg, kE, kN, kC);
    fill_kernel<<<CDIV((int)kncPad, TB), TB, 0, stream>>>(bufPad, (int)kncPad, 0.0f);
    combine_kernel<<<CDIV((int)knc, TB), TB, 0, stream>>>(
        agg, rootb, b2, bufPad, kN, kC, kCPad, (int)knc);
    // t = 1 : clean unguarded WMMA path with lda = 20
    gemm(bufPad, (size_t)kN * kCPad, kCPad, w2, bufA, kN, kC, kC);  // h1 = out[k] @ w2[k]
    fill_kernel<<<CDIV((int)knc, TB), TB, 0, stream>>>(agg, (int)knc, 0.0f);
    scatter_scalar_kernel<<<dim3(CDIV(kE * kC, TB), kK), TB, 0, stream>>>(
        bufA, ei, nrm, agg, kE, kN, kC);
    combine_kernel<<<CDIV((int)knc, TB), TB, 0, stream>>>(
        agg, rootb, b2, bufB, kN, kC, kC, (int)knc);

    // mean over stacks + log_softmax -> d_out
    mean_logsoftmax_kernel<<<CDIV(kN, TB), TB, 0, stream>>>(bufB, out, kN, kC);
}